// ThresholdLoss_60670708023920
// MI455X (gfx1250) — compile-verified
//
#include <hip/hip_runtime.h>
#include <hip/hip_bf16.h>
#include <math.h>

// ---------------------------------------------------------------------------
// CDNA5 wave32 reduction via V_WMMA_F32_16X16X4_F32.
// A-tile (16x4 f32, 2 VGPRs/lane): lane L(0-15) supplies A[L][0]=x, A[L][1]=0;
// lane L(16-31) supplies A[L-16][2]=x, A[L-16][3]=0.  B = all-ones 4x16
// (layout independent).  D[m][n] = x_m + x_{m+16} for every column n.
// C/D layout: lanes 0-15 VGPR r hold M=r, lanes 16-31 hold M=8+r.  Summing the
// 8 accumulator VGPRs per lane: lane0 = sum(rows 0..7), lane16 = sum(rows
// 8..15); their sum is the exact f32 wave total.
// ---------------------------------------------------------------------------
typedef __attribute__((ext_vector_type(2))) float v2f;
typedef __attribute__((ext_vector_type(4))) float v4f;
typedef __attribute__((ext_vector_type(8))) float v8f;

__device__ __forceinline__ float wave_sum_wmma(float x) {
    v2f a; a[0] = x;    a[1] = 0.0f;
    v2f b; b[0] = 1.0f; b[1] = 1.0f;
    v8f c = {};
    c = __builtin_amdgcn_wmma_f32_16x16x4_f32(false, a, false, b,
                                              (short)0, c, false, false);
    float s = ((c[0] + c[1]) + (c[2] + c[3])) + ((c[4] + c[5]) + (c[6] + c[7]));
    return __shfl(s, 0, 32) + __shfl(s, 16, 32);
}

// ---------------------------------------------------------------------------
// Kernel 1: per-block partials over targets: {sum, sumsq, min, max}
// ---------------------------------------------------------------------------
__global__ __launch_bounds__(256) void k_stats(const float* __restrict__ t,
                                               int n, float4* __restrict__ part) {
    __shared__ float ls[8], lq[8], lmn[8], lmx[8];
    const int  tid    = threadIdx.x;
    const long gid    = (long)blockIdx.x * blockDim.x + tid;
    const long stride = (long)gridDim.x * blockDim.x;
    const long n4     = (long)n >> 2;

    float s0 = 0.0f, s1 = 0.0f, s2 = 0.0f, s3 = 0.0f;
    float q0 = 0.0f, q1 = 0.0f, q2 = 0.0f, q3 = 0.0f;
    float mn = __builtin_inff(), mx = -__builtin_inff();

    const v4f* t4 = (const v4f*)t;
    long i = gid;
    // 4x unrolled: 4 independent b128 loads in flight per iteration
    for (; i + 3 * stride < n4; i += 4 * stride) {
        v4f a = t4[i];
        v4f b = t4[i + stride];
        v4f c = t4[i + 2 * stride];
        v4f d = t4[i + 3 * stride];
        #pragma unroll
        for (int k = 0; k < 4; ++k) {
            s0 += a[k]; q0 = fmaf(a[k], a[k], q0); mn = fminf(mn, a[k]); mx = fmaxf(mx, a[k]);
            s1 += b[k]; q1 = fmaf(b[k], b[k], q1); mn = fminf(mn, b[k]); mx = fmaxf(mx, b[k]);
            s2 += c[k]; q2 = fmaf(c[k], c[k], q2); mn = fminf(mn, c[k]); mx = fmaxf(mx, c[k]);
            s3 += d[k]; q3 = fmaf(d[k], d[k], q3); mn = fminf(mn, d[k]); mx = fmaxf(mx, d[k]);
        }
    }
    for (; i < n4; i += stride) {
        v4f a = t4[i];
        #pragma unroll
        for (int k = 0; k < 4; ++k) {
            s0 += a[k]; q0 = fmaf(a[k], a[k], q0); mn = fminf(mn, a[k]); mx = fmaxf(mx, a[k]);
        }
    }
    for (long j = (n4 << 2) + gid; j < (long)n; j += stride) {  // scalar tail
        float v = t[j];
        s0 += v; q0 = fmaf(v, v, q0); mn = fminf(mn, v); mx = fmaxf(mx, v);
    }

    float s = (s0 + s1) + (s2 + s3);
    float q = (q0 + q1) + (q2 + q3);

    // wave32 reductions
    float wsum = wave_sum_wmma(s);
    float wsq  = wave_sum_wmma(q);
    for (int m = 16; m >= 1; m >>= 1) {
        mn = fminf(mn, __shfl_xor(mn, m, 32));
        mx = fmaxf(mx, __shfl_xor(mx, m, 32));
    }

    const int lane = tid & 31, wid = tid >> 5;
    if (lane == 0) { ls[wid] = wsum; lq[wid] = wsq; lmn[wid] = mn; lmx[wid] = mx; }
    __syncthreads();
    if (tid == 0) {
        float S = 0.0f, Q = 0.0f, MN = __builtin_inff(), MX = -__builtin_inff();
        for (int k = 0; k < 8; ++k) {
            S += ls[k]; Q += lq[k]; MN = fminf(MN, lmn[k]); MX = fmaxf(MX, lmx[k]);
        }
        part[blockIdx.x] = make_float4(S, Q, MN, MX);
    }
}

// ---------------------------------------------------------------------------
// Kernel 2: reduce partials -> scalars {gmin, gmax, m1, m2}
// ---------------------------------------------------------------------------
__global__ __launch_bounds__(256) void k_stats_fin(const float4* __restrict__ part,
                                                   int nb, long n,
                                                   float* __restrict__ sc) {
    __shared__ float ls[8], lq[8], lmn[8], lmx[8];
    const int tid = threadIdx.x;
    float s = 0.0f, q = 0.0f;
    float mn = __builtin_inff(), mx = -__builtin_inff();
    for (int i = tid; i < nb; i += 256) {
        float4 p = part[i];
        s += p.x; q += p.y; mn = fminf(mn, p.z); mx = fmaxf(mx, p.w);
    }
    float wsum = wave_sum_wmma(s);
    float wsq  = wave_sum_wmma(q);
    for (int m = 16; m >= 1; m >>= 1) {
        mn = fminf(mn, __shfl_xor(mn, m, 32));
        mx = fmaxf(mx, __shfl_xor(mx, m, 32));
    }
    const int lane = tid & 31, wid = tid >> 5;
    if (lane == 0) { ls[wid] = wsum; lq[wid] = wsq; lmn[wid] = mn; lmx[wid] = mx; }
    __syncthreads();
    if (tid == 0) {
        float S = 0.0f, Q = 0.0f, MN = __builtin_inff(), MX = -__builtin_inff();
        for (int k = 0; k < 8; ++k) {
            S += ls[k]; Q += lq[k]; MN = fminf(MN, lmn[k]); MX = fmaxf(MX, lmx[k]);
        }
        float fn   = (float)n;
        float mean = S / fn;
        float var  = (Q - S * S / fn) / (float)(n - 1);   // ddof = 1
        float sd   = sqrtf(var);
        sc[0] = MN;                // gmin
        sc[1] = MX;                // gmax
        sc[2] = mean + sd;         // m1
        sc[3] = mean + 3.0f * sd;  // m2
    }
}

// ---------------------------------------------------------------------------
// Kernel 3: weighted-L1 partial sums.
// targets: default (RT) loads -> should hit the 192MB L2 from pass 1.
// inputs:  read exactly once  -> non-temporal so they don't evict targets.
// ---------------------------------------------------------------------------
__device__ __forceinline__ float w_of(float g, float m1, float m2, float gmin,
                                      float inv_lo, float inv_mi, float inv_hi) {
    return (g <= m1) ? (g - gmin) * inv_lo
         : (g <= m2) ? fmaf(g - m1, inv_mi, 0.2f)
                     : fmaf(g - m2, inv_hi, 0.7f);
}

__global__ __launch_bounds__(256) void k_wloss(const float* __restrict__ x,
                                               const float* __restrict__ t,
                                               int n, const float* __restrict__ sc,
                                               float* __restrict__ part) {
    __shared__ float ls[8];
    const float gmin = sc[0], gmax = sc[1], m1 = sc[2], m2 = sc[3];
    const float inv_lo = 0.2f / (m1 - gmin);
    const float inv_mi = 0.5f / (m2 - m1);
    const float inv_hi = 0.3f / (gmax - m2);

    const int  tid    = threadIdx.x;
    const long gid    = (long)blockIdx.x * blockDim.x + tid;
    const long stride = (long)gridDim.x * blockDim.x;
    const long n4     = (long)n >> 2;

    float acc0 = 0.0f, acc1 = 0.0f, acc2 = 0.0f, acc3 = 0.0f;
    const v4f* x4 = (const v4f*)x;
    const v4f* t4 = (const v4f*)t;

    long i = gid;
    // 4x unrolled: 8 independent b128 loads in flight per iteration
    for (; i + 3 * stride < n4; i += 4 * stride) {
        v4f ta = t4[i];
        v4f tb = t4[i + stride];
        v4f tc = t4[i + 2 * stride];
        v4f td = t4[i + 3 * stride];
        v4f xa = __builtin_nontemporal_load(x4 + i);
        v4f xb = __builtin_nontemporal_load(x4 + i + stride);
        v4f xc = __builtin_nontemporal_load(x4 + i + 2 * stride);
        v4f xd = __builtin_nontemporal_load(x4 + i + 3 * stride);

        #pragma unroll
        for (int k = 0; k < 4; ++k) {
            acc0 = fmaf(fabsf(ta[k] - xa[k]), w_of(ta[k], m1, m2, gmin, inv_lo, inv_mi, inv_hi), acc0);
            acc1 = fmaf(fabsf(tb[k] - xb[k]), w_of(tb[k], m1, m2, gmin, inv_lo, inv_mi, inv_hi), acc1);
            acc2 = fmaf(fabsf(tc[k] - xc[k]), w_of(tc[k], m1, m2, gmin, inv_lo, inv_mi, inv_hi), acc2);
            acc3 = fmaf(fabsf(td[k] - xd[k]), w_of(td[k], m1, m2, gmin, inv_lo, inv_mi, inv_hi), acc3);
        }
    }
    for (; i < n4; i += stride) {
        v4f ta = t4[i];
        v4f xa = __builtin_nontemporal_load(x4 + i);
        #pragma unroll
        for (int k = 0; k < 4; ++k) {
            acc0 = fmaf(fabsf(ta[k] - xa[k]), w_of(ta[k], m1, m2, gmin, inv_lo, inv_mi, inv_hi), acc0);
        }
    }
    for (long j = (n4 << 2) + gid; j < (long)n; j += stride) {  // scalar tail
        float g = t[j];
        acc0 = fmaf(fabsf(g - x[j]), w_of(g, m1, m2, gmin, inv_lo, inv_mi, inv_hi), acc0);
    }

    float acc = (acc0 + acc1) + (acc2 + acc3);
    float wsum = wave_sum_wmma(acc);
    const int lane = tid & 31, wid = tid >> 5;
    if (lane == 0) ls[wid] = wsum;
    __syncthreads();
    if (tid == 0) {
        float S = 0.0f;
        for (int k = 0; k < 8; ++k) S += ls[k];
        part[blockIdx.x] = S;
    }
}

// ---------------------------------------------------------------------------
// Kernel 4: final mean -> d_out[0]
// ---------------------------------------------------------------------------
__global__ __launch_bounds__(256) void k_final(const float* __restrict__ part,
                                               int nb, long n,
                                               float* __restrict__ out) {
    __shared__ float ls[8];
    const int tid = threadIdx.x;
    float s = 0.0f;
    for (int i = tid; i < nb; i += 256) s += part[i];
    float wsum = wave_sum_wmma(s);
    const int lane = tid & 31, wid = tid >> 5;
    if (lane == 0) ls[wid] = wsum;
    __syncthreads();
    if (tid == 0) {
        float S = 0.0f;
        for (int k = 0; k < 8; ++k) S += ls[k];
        out[0] = S / (float)n;
    }
}

// ---------------------------------------------------------------------------
extern "C" void kernel_launch(void* const* d_in, const int* in_sizes, int n_in,
                              void* d_out, int out_size, void* d_ws, size_t ws_size,
                              hipStream_t stream) {
    (void)n_in; (void)out_size;
    const float* x = (const float*)d_in[0];  // inputs
    const float* t = (const float*)d_in[1];  // targets
    float* out = (float*)d_out;
    const int n = in_sizes[0];               // 4096*4096 = 16,777,216

    // workspace layout: [float4 part1[nb]] [float sc[4]] [float part3[nb]]
    int nb = 1024;
    while (nb > 1 && ((size_t)nb * 16 + 16 + (size_t)nb * 4) > ws_size) nb >>= 1;

    char*   ws    = (char*)d_ws;
    float4* part1 = (float4*)ws;
    float*  sc    = (float*)(ws + (size_t)nb * 16);
    float*  part3 = (float*)(ws + (size_t)nb * 16 + 16);

    k_stats    <<<nb, 256, 0, stream>>>(t, n, part1);
    k_stats_fin<<<1,  256, 0, stream>>>(part1, nb, (long)n, sc);
    k_wloss    <<<nb, 256, 0, stream>>>(x, t, n, sc, part3);
    k_final    <<<1,  256, 0, stream>>>(part3, nb, (long)n, out);
}